// DHHPTransform_28673201668759
// MI455X (gfx1250) — compile-verified
//
#include <hip/hip_runtime.h>

typedef float v2f __attribute__((ext_vector_type(2)));
typedef float v8f __attribute__((ext_vector_type(8)));

constexpr int Bb    = 8;
constexpr int N     = 8192;
constexpr int D     = 256;
constexpr int H     = N / 2;      // 4096
constexpr int LOG2H = 12;
constexpr int NG    = N - 1;
constexpr int BR    = 32;         // output rows per block
constexpr int WIN   = BR + 4;     // x window rows (halo of 2 each side)
constexpr int XS    = D + 8;      // padded LDS row stride (floats) -> conflict-free halves

// Upper sweep: y[i] = P*x[i-1] + Q*x[i] + R*x[i+1]
__device__ __forceinline__ void pqr_u(const float* gii, const float* gij,
                                      const float* gji, const float* gjj,
                                      int i, float& P, float& Q, float& R) {
  if (i < 0 || i >= N) { P = 0.f; Q = 0.f; R = 0.f; return; }
  if (i == 0)          { P = 0.f; Q = gii[0]; R = gij[0]; return; }
  if (i == N - 1)      { P = gji[N - 2]; Q = gjj[N - 2]; R = 0.f; return; }
  float jj = gjj[i - 1];
  P = gji[i - 1];
  Q = jj * gii[i];
  R = jj * gij[i];
}

// Lower sweep: z[r] = L*y[r-1] + M*y[r] + Hh*y[r+1]
__device__ __forceinline__ void lmh_l(const float* gii, const float* gij,
                                      const float* gji, const float* gjj,
                                      int r, float& L, float& M, float& Hh) {
  if (r == 0)     { L = 0.f; M = gii[0]; Hh = gij[0]; return; }
  if (r == N - 1) { L = gji[N - 2]; M = gjj[N - 2]; Hh = 0.f; return; }
  float ii = gii[r];
  L  = ii * gji[r - 1];
  M  = ii * gjj[r - 1];
  Hh = gij[r];
}

// select weight by dynamic tap index (0..4), 0 outside -> v_cndmask chain
__device__ __forceinline__ float wsel(int j, float w0, float w1, float w2,
                                      float w3, float w4) {
  float r = 0.f;
  r = (j == 0) ? w0 : r;
  r = (j == 1) ? w1 : r;
  r = (j == 2) ? w2 : r;
  r = (j == 3) ? w3 : r;
  r = (j == 4) ? w4 : r;
  return r;
}

__global__ __launch_bounds__(256, 1) void dhhp_fused(
    const float* __restrict__ in,
    const float* __restrict__ glii, const float* __restrict__ glij,
    const float* __restrict__ glji, const float* __restrict__ gljj,
    const float* __restrict__ guii, const float* __restrict__ guij,
    const float* __restrict__ guji, const float* __restrict__ gujj,
    const float* __restrict__ diag, const int* __restrict__ trp,
    float* __restrict__ out) {
  __shared__ __align__(16) float buf[WIN * XS];   // x window, reused for z
  __shared__ float wts[5 * BR];

  const int t    = threadIdx.x;
  const int bb   = blockIdx.y;
  const int rblk = blockIdx.x * BR;
  const int tr   = trp[0];

  const float* xin = in   + (size_t)bb * N * D;
  const float* dgb = diag + (size_t)bb * N;

  // ---------- stage x window into LDS (permute / diag-scale on the fly) ----
  {
    const int c4 = (t & 63) * 4;
    for (int rr = (t >> 6); rr < WIN; rr += 4) {
      int gr = rblk - 2 + rr;
      gr = gr < 0 ? 0 : (gr > N - 1 ? N - 1 : gr);   // clamped halo: weights are 0 there
      int srow = tr ? (((gr & (H - 1)) << 1) + (gr >> LOG2H)) : gr;
      float4 v = *(const float4*)(xin + (size_t)srow * D + c4);
      if (!tr) {
        float dg = dgb[gr];
        v.x *= dg; v.y *= dg; v.z *= dg; v.w *= dg;
      }
      *(float4*)(&buf[rr * XS + c4]) = v;
    }
  }

  // ---------- fused 5-tap weights per output row ----------
  if (t < BR) {
    const float* lii = glii + (size_t)bb * NG;
    const float* lij = glij + (size_t)bb * NG;
    const float* lji = glji + (size_t)bb * NG;
    const float* ljj = gljj + (size_t)bb * NG;
    const float* uii = guii + (size_t)bb * NG;
    const float* uij = guij + (size_t)bb * NG;
    const float* uji = guji + (size_t)bb * NG;
    const float* ujj = gujj + (size_t)bb * NG;

    int r = rblk + t;
    float Pm, Qm, Rm, P0, Q0, R0, Pp, Qp, Rp, Lr, Mr, Hr;
    pqr_u(uii, uij, uji, ujj, r - 1, Pm, Qm, Rm);
    pqr_u(uii, uij, uji, ujj, r,     P0, Q0, R0);
    pqr_u(uii, uij, uji, ujj, r + 1, Pp, Qp, Rp);
    lmh_l(lii, lij, lji, ljj, r, Lr, Mr, Hr);
    wts[0 * BR + t] = Lr * Pm;
    wts[1 * BR + t] = Lr * Qm + Mr * P0;
    wts[2 * BR + t] = Lr * Rm + Mr * Q0 + Hr * Pp;
    wts[3 * BR + t] = Mr * R0 + Hr * Qp;
    wts[4 * BR + t] = Hr * Rp;
  }
  __syncthreads();

  // ---------- banded operator as 5 chained V_WMMA_F32_16X16X4_F32 ----------
  const int wave = t >> 5, lane = t & 31;
  const int hh = lane >> 4, nl = lane & 15;
  const int rt = wave >> 2;                 // row sub-tile (0..1)
  const int rowin = rt * 16 + nl;
  const float w0 = wts[0 * BR + rowin];
  const float w1 = wts[1 * BR + rowin];
  const float w2 = wts[2 * BR + rowin];
  const float w3 = wts[3 * BR + rowin];
  const float w4 = wts[4 * BR + rowin];

  v8f acc[4];
  for (int k = 0; k < 4; ++k) {
    const int n0 = ((wave & 3) * 4 + k) * 16;   // column sub-tile base
    v8f c = {0.f, 0.f, 0.f, 0.f, 0.f, 0.f, 0.f, 0.f};
#pragma unroll
    for (int q = 0; q < 5; ++q) {
      int kb = 4 * q + 2 * hh;                  // band column for VGPR0 of this lane half
      v2f bm;                                   // B 4x16: VGPR0 -> K=2*hh, VGPR1 -> K=2*hh+1
      bm.x = buf[(rt * 16 + kb + 0) * XS + n0 + nl];
      bm.y = buf[(rt * 16 + kb + 1) * XS + n0 + nl];
      int j0 = kb - nl;                         // tap index = bandcol - row
      v2f am;                                   // A 16x4: M=nl, K=2*hh (+1 for VGPR1)
      am.x = wsel(j0,     w0, w1, w2, w3, w4);
      am.y = wsel(j0 + 1, w0, w1, w2, w3, w4);
      c = __builtin_amdgcn_wmma_f32_16x16x4_f32(false, am, false, bm,
                                                (short)0, c, false, false);
    }
    acc[k] = c;
  }
  __syncthreads();   // all LDS x reads done before reuse

  // ---------- scatter D-tiles back to LDS (z) ----------
  for (int k = 0; k < 4; ++k) {
    const int n0 = ((wave & 3) * 4 + k) * 16;
#pragma unroll
    for (int g = 0; g < 8; ++g) {
      buf[(rt * 16 + g + 8 * hh) * XS + n0 + nl] = acc[k][g];
    }
  }
  __syncthreads();

  // ---------- fully coalesced stores (diag-scale or inverse permutation) ----
  {
    const int c4 = (t & 63) * 4;
    float* outb = out + (size_t)bb * N * D;
    for (int rr = (t >> 6); rr < BR; rr += 4) {
      int r = rblk + rr;
      float4 v = *(const float4*)(&buf[rr * XS + c4]);
      int orow;
      if (tr) {
        float dg = dgb[r];
        v.x *= dg; v.y *= dg; v.z *= dg; v.w *= dg;
        orow = r;
      } else {
        orow = ((r & 1) << LOG2H) + (r >> 1);   // inverse of stride-2 permutation
      }
      *(float4*)(outb + (size_t)orow * D + c4) = v;
    }
  }
}

extern "C" void kernel_launch(void* const* d_in, const int* in_sizes, int n_in,
                              void* d_out, int out_size, void* d_ws, size_t ws_size,
                              hipStream_t stream) {
  (void)in_sizes; (void)n_in; (void)out_size; (void)d_ws; (void)ws_size;
  const float* in   = (const float*)d_in[0];
  const float* glii = (const float*)d_in[1];
  const float* glij = (const float*)d_in[2];
  const float* glji = (const float*)d_in[3];
  const float* gljj = (const float*)d_in[4];
  const float* guii = (const float*)d_in[5];
  const float* guij = (const float*)d_in[6];
  const float* guji = (const float*)d_in[7];
  const float* gujj = (const float*)d_in[8];
  const float* diag = (const float*)d_in[9];
  const int*   trp  = (const int*)d_in[10];
  float* out = (float*)d_out;

  dim3 grid(N / BR, Bb);
  dhhp_fused<<<grid, 256, 0, stream>>>(in, glii, glij, glji, gljj,
                                       guii, guij, guji, gujj, diag, trp, out);
}